// TransformerAttention_25486335935208
// MI455X (gfx1250) — compile-verified
//
#include <hip/hip_runtime.h>
#include <hip/hip_bf16.h>
#include <math.h>

// Problem constants (from reference)
#define B_  2
#define S_  2048
#define D_  1024
#define H_  16
#define DK_ 64
// H_*DK_ == D_ == 1024 (row stride for Q/K/V/ctx)

typedef __attribute__((ext_vector_type(2))) float v2f;
typedef __attribute__((ext_vector_type(8))) float v8f;

// Dense fp32 WMMA: D = A(16x4) * B(4x16) + C(16x16)
// 8-arg form: (neg_a, A, neg_b, B, c_mod, C, reuse_a, reuse_b)
static __device__ __forceinline__ v8f wmma_f32(v2f a, v2f b, v8f c) {
  return __builtin_amdgcn_wmma_f32_16x16x4_f32(false, a, false, b, (short)0, c,
                                               false, false);
}

// Async global->LDS copy (CDNA5, tracked by ASYNCcnt). VDST VGPR holds the
// LDS byte address (= low 32 bits of the generic shared pointer, per the
// flat-address aperture rules); VADDR pair holds the global address.
static __device__ __forceinline__ void async_load_b128(uint32_t lds_addr,
                                                       const float* gaddr) {
  asm volatile("global_load_async_to_lds_b128 %0, %1, off"
               :
               : "v"(lds_addr), "v"(gaddr)
               : "memory");
}
static __device__ __forceinline__ void wait_asynccnt0() {
  asm volatile("s_wait_asynccnt 0x0" ::: "memory");
}

// ---------------------------------------------------------------------------
// Kernel 1: relative-position bucket LUT for rel = mem - ctx in [-2047, 2047]
// lut[rel + 2047] = bucket index into rel_bias[NUM_BUCKETS][H]
// ---------------------------------------------------------------------------
__global__ void t5_bucket_lut_kernel(int* __restrict__ lut) {
  int i = blockIdx.x * blockDim.x + threadIdx.x;
  if (i >= 4095) return;
  int rel = i - 2047;            // mem - ctx
  int n = -rel;                  // ctx - mem
  int ret = (n < 0) ? 16 : 0;    // num_buckets//2 = 16
  n = (n < 0) ? -n : n;
  int bucket;
  if (n < 8) {                   // max_exact = 8
    bucket = n;
  } else {
    // 8 + int(log(n/8)/log(128/8) * 8), clamped to 15
    float v = logf((float)n * 0.125f) * (8.0f / logf(16.0f));
    int vi = (int)v;             // truncation toward zero (v >= 0)
    bucket = 8 + (vi < 7 ? vi : 7);
  }
  lut[i] = ret + bucket;
}

// ---------------------------------------------------------------------------
// Kernel 2: GEMM  C[M,N] = A[M,K] * W[N,K]^T   (all fp32, WMMA f32 16x16x4)
// Block: 128 threads (4 waves). Block tile 64(M) x 64(N), K staged by 32,
// double-buffered in LDS via global_load_async_to_lds_b128 (ASYNCcnt).
// Wave w computes rows [mBase + 16w, +16), all 64 N columns (4 accum tiles).
// ---------------------------------------------------------------------------
#define LDS_STRIDE 40  // 40 floats/row: 160B rows -> 16B-aligned b128 stores,
                       // even offsets -> 8B-aligned v2f fragment reads
__global__ __launch_bounds__(128) void gemm_nt_wmma_f32(
    const float* __restrict__ A, const float* __restrict__ W,
    float* __restrict__ C, int M, int N, int K) {
  __shared__ __align__(16) float As[2][64 * LDS_STRIDE];
  __shared__ __align__(16) float Ws[2][64 * LDS_STRIDE];

  const int tid  = threadIdx.x;
  const int wave = tid >> 5;
  const int lane = tid & 31;
  const int l15  = lane & 15;
  const int hi   = lane >> 4;          // 0: K=0,1 half, 1: K=2,3 half
  const int mBase = blockIdx.x * 64;
  const int nBase = blockIdx.y * 64;
  const int mSub  = wave * 16;

  v8f acc[4] = {};

  const int row = tid >> 3;            // 0..15
  const int c4  = (tid & 7) * 4;       // 0..28 step 4

  // Issue async loads of the 64x32 A and W tiles for chunk starting at k0
  // into buffer `buf`. 128 threads x 8 x b128 = both tiles.
  auto issue_tiles = [&](int buf, int k0) {
#pragma unroll
    for (int rr = 0; rr < 4; ++rr) {
      const int r = row + rr * 16;
      async_load_b128((uint32_t)(uintptr_t)&As[buf][r * LDS_STRIDE + c4],
                      &A[(long)(mBase + r) * K + k0 + c4]);
      async_load_b128((uint32_t)(uintptr_t)&Ws[buf][r * LDS_STRIDE + c4],
                      &W[(long)(nBase + r) * K + k0 + c4]);
    }
  };

  const int nchunks = K / 32;
  issue_tiles(0, 0);

  for (int c = 0; c < nchunks; ++c) {
    const int buf = c & 1;
    // My async loads done; barrier makes every wave's tile writes visible.
    wait_asynccnt0();
    __syncthreads();
    // Prefetch next chunk into the other buffer while we compute.
    // (Safe: all waves finished reading that buffer before this barrier --
    // their ds_load results were consumed by WMMAs behind s_wait_dscnt.)
    if (c + 1 < nchunks) issue_tiles(buf ^ 1, (c + 1) * 32);

    const float* as = As[buf];
    const float* ws = Ws[buf];
    // --- 8 k-steps of 4, 4 N-tiles each ---
#pragma unroll
    for (int kk = 0; kk < 8; ++kk) {
      const float* ap = &as[(mSub + l15) * LDS_STRIDE + kk * 4 + hi * 2];
      v2f a; a.x = ap[0]; a.y = ap[1];
      const float* wp = &ws[l15 * LDS_STRIDE + kk * 4 + hi * 2];
#pragma unroll
      for (int t = 0; t < 4; ++t) {
        const float* wpt = wp + t * 16 * LDS_STRIDE;
        v2f b; b.x = wpt[0]; b.y = wpt[1];
        acc[t] = wmma_f32(a, b, acc[t]);
      }
    }
  }

  // --- epilogue: C-layout = VGPR r -> row (r + 8*hi), lane -> column ---
#pragma unroll
  for (int t = 0; t < 4; ++t) {
#pragma unroll
    for (int r = 0; r < 8; ++r) {
      C[(long)(mBase + mSub + r + 8 * hi) * N + nBase + 16 * t + l15] =
          acc[t][r];
    }
  }
}

// ---------------------------------------------------------------------------
// Kernel 3: flash attention with T5 relative-position bias.
// Q,K,V,ctx layout: [B, S, H*DK] (row stride 1024). One wave per 16-row
// query tile; 8 waves per block. Streaming softmax, scores never hit memory.
// ---------------------------------------------------------------------------
__global__ __launch_bounds__(256) void flash_attn_t5_kernel(
    const float* __restrict__ Q, const float* __restrict__ Km,
    const float* __restrict__ Vm, const float* __restrict__ relBias,
    const int* __restrict__ lut, float* __restrict__ ctx) {
  __shared__ __align__(16) float pLds[8 * 16 * 18];  // per-wave 16x18 pad

  const int tid  = threadIdx.x;
  const int wave = tid >> 5;
  const int lane = tid & 31;
  const int l15  = lane & 15;
  const int hi   = lane >> 4;
  const int h    = blockIdx.y;
  const int b    = blockIdx.z;
  const int qBase = (blockIdx.x * 8 + wave) * 16;
  float* pl = &pLds[wave * 16 * 18];

  const long RS = (long)H_ * DK_;  // 1024

  // Preload Q fragments for all 16 k-chunks of DK=64 (A-matrix layout:
  // lanes 0-15 hold K=4kk+0,1 ; lanes 16-31 hold K=4kk+2,3, row M = l15).
  const float* qRow = Q + ((long)b * S_ + qBase + l15) * RS + h * DK_;
  v2f qf[16];
#pragma unroll
  for (int kk = 0; kk < 16; ++kk) {
    qf[kk].x = qRow[kk * 4 + hi * 2];
    qf[kk].y = qRow[kk * 4 + hi * 2 + 1];
  }

  float mi[8], li[8];
#pragma unroll
  for (int r = 0; r < 8; ++r) { mi[r] = -INFINITY; li[r] = 0.0f; }
  v8f acc[4] = {};

  for (int kt = 0; kt < S_ / 16; ++kt) {
    const int kBase = kt * 16;

    // ---- scores = Q * K^T over DK=64 (16 WMMA k-steps) ----
    v8f sc = {};
    const float* kRow = Km + ((long)b * S_ + kBase + l15) * RS + h * DK_;
#pragma unroll
    for (int kk = 0; kk < 16; ++kk) {
      v2f kf;  // B[d][k'] = K[kBase+k'][d]; column n = l15, rows d = 4kk+...
      kf.x = kRow[kk * 4 + hi * 2];
      kf.y = kRow[kk * 4 + hi * 2 + 1];
      sc = wmma_f32(qf[kk], kf, sc);
    }

    // ---- bias add + online softmax (per C-layout row r) ----
    float p[8];
    const int kpos = kBase + l15;
#pragma unroll
    for (int r = 0; r < 8; ++r) {
      const int qpos = qBase + r + 8 * hi;
      const int bucket = lut[kpos - qpos + 2047];
      const float s = sc[r] + relBias[bucket * H_ + h];
      float m = s;  // row-max across the 16 lanes of this half
      m = fmaxf(m, __shfl_xor(m, 1, 32));
      m = fmaxf(m, __shfl_xor(m, 2, 32));
      m = fmaxf(m, __shfl_xor(m, 4, 32));
      m = fmaxf(m, __shfl_xor(m, 8, 32));
      const float mnew  = fmaxf(mi[r], m);
      const float alpha = __expf(mi[r] - mnew);
      float pv = __expf(s - mnew);
      float rs = pv;
      rs += __shfl_xor(rs, 1, 32);
      rs += __shfl_xor(rs, 2, 32);
      rs += __shfl_xor(rs, 4, 32);
      rs += __shfl_xor(rs, 8, 32);
      li[r] = li[r] * alpha + rs;
      mi[r] = mnew;
      p[r]  = pv;
      acc[0][r] *= alpha; acc[1][r] *= alpha;
      acc[2][r] *= alpha; acc[3][r] *= alpha;
    }

    // ---- transpose P (C-layout -> A-layout) through per-wave LDS ----
#pragma unroll
    for (int r = 0; r < 8; ++r)
      pl[(r + 8 * hi) * 18 + l15] = p[r];
    asm volatile("s_wait_dscnt 0" ::: "memory");

    // ---- acc += P(16x16) * V(16x64), K-dim 16 in 4 steps ----
    const float* vBase = Vm + ((long)b * S_ + kBase) * RS + h * DK_;
#pragma unroll
    for (int kk = 0; kk < 4; ++kk) {
      const float* pp = &pl[l15 * 18 + kk * 4 + hi * 2];
      v2f pf; pf.x = pp[0]; pf.y = pp[1];
      const float* vr = vBase + (long)(kk * 4 + hi * 2) * RS;
#pragma unroll
      for (int t = 0; t < 4; ++t) {
        v2f vf;  // B[k'][d]: rows k' = 4kk + {2hi, 2hi+1}, column d = 16t+l15
        vf.x = vr[t * 16 + l15];
        vf.y = vr[RS + t * 16 + l15];
        acc[t] = wmma_f32(pf, vf, acc[t]);
      }
    }
    asm volatile("s_wait_dscnt 0" ::: "memory");  // WAR guard on pl
  }

  // ---- normalize and store context [B, S, H*DK] ----
#pragma unroll
  for (int r = 0; r < 8; ++r) {
    const float inv = 1.0f / li[r];
    float* orow = ctx + ((long)b * S_ + qBase + r + 8 * hi) * RS + h * DK_;
#pragma unroll
    for (int t = 0; t < 4; ++t)
      orow[t * 16 + l15] = acc[t][r] * inv;
  }
}

// ---------------------------------------------------------------------------
// Host-side launcher
// ---------------------------------------------------------------------------
extern "C" void kernel_launch(void* const* d_in, const int* in_sizes, int n_in,
                              void* d_out, int out_size, void* d_ws,
                              size_t ws_size, hipStream_t stream) {
  const float* x  = (const float*)d_in[0];  // [B,S,D]
  const float* qw = (const float*)d_in[1];  // [H*DK, D]
  const float* kw = (const float*)d_in[2];
  const float* vw = (const float*)d_in[3];
  const float* ow = (const float*)d_in[4];  // [D, H*DK]
  const float* rb = (const float*)d_in[5];  // [32, H]
  float* out = (float*)d_out;               // [B,S,D]

  const long NM = (long)B_ * S_ * (H_ * DK_);  // 4096*1024 elements
  float* qBuf = (float*)d_ws;
  float* kBuf = qBuf + NM;
  float* vBuf = kBuf + NM;
  float* ctxB = vBuf + NM;
  int*   lut  = (int*)(ctxB + NM);

  // 1) bucket LUT
  t5_bucket_lut_kernel<<<dim3(16), dim3(256), 0, stream>>>(lut);

  // 2) Q/K/V projections: [4096,1024] = x[4096,1024] * W[1024,1024]^T
  const int M = B_ * S_, N = H_ * DK_, K = D_;
  dim3 gg(M / 64, N / 64);
  gemm_nt_wmma_f32<<<gg, 128, 0, stream>>>(x, qw, qBuf, M, N, K);
  gemm_nt_wmma_f32<<<gg, 128, 0, stream>>>(x, kw, kBuf, M, N, K);
  gemm_nt_wmma_f32<<<gg, 128, 0, stream>>>(x, vw, vBuf, M, N, K);

  // 3) flash attention with T5 bias
  flash_attn_t5_kernel<<<dim3(S_ / 128, H_, B_), 256, 0, stream>>>(
      qBuf, kBuf, vBuf, rb, lut, ctxB);

  // 4) output projection: out = ctx * o_w^T
  dim3 go(M / 64, D_ / 64);
  gemm_nt_wmma_f32<<<go, 128, 0, stream>>>(ctxB, ow, out, M, D_, K);
}